// TestNet_30502857736792
// MI455X (gfx1250) — compile-verified
//
#include <hip/hip_runtime.h>
#include <hip/hip_bf16.h>
#include <cstdint>
#include <cstddef>

typedef float v2f __attribute__((ext_vector_type(2)));
typedef float v8f __attribute__((ext_vector_type(8)));

#define TPB 256
static inline int cdiv(long long a, long long b) { return (int)((a + b - 1) / b); }

// ---------------- degree / edge-norm kernels ----------------

__global__ void deg_kernel(const int* __restrict__ src, const int* __restrict__ dst,
                           float* __restrict__ degc, float* __restrict__ degg, int E) {
  int e = blockIdx.x * blockDim.x + threadIdx.x;
  if (e >= E) return;
  atomicAdd(&degc[src[e]], 1.0f);   // Cheb: segment_sum(ones, src)
  atomicAdd(&degg[dst[e]], 1.0f);   // GCN: in-degree (self-loop added later)
}

__global__ void dinv_kernel(const float* __restrict__ degc, const float* __restrict__ degg,
                            float* __restrict__ dinvc, float* __restrict__ dinvg, int N) {
  int i = blockIdx.x * blockDim.x + threadIdx.x;
  if (i >= N) return;
  float dc = degc[i];
  dinvc[i] = dc > 0.f ? rsqrtf(fmaxf(dc, 1e-12f)) : 0.f;
  dinvg[i] = rsqrtf(degg[i] + 1.0f);   // +1 self loop; always > 0
}

__global__ void edgew_kernel(const int* __restrict__ src, const int* __restrict__ dst,
                             const float* __restrict__ dinvc, const float* __restrict__ dinvg,
                             float* __restrict__ wch, float* __restrict__ wgc, int E) {
  int e = blockIdx.x * blockDim.x + threadIdx.x;
  if (e >= E) return;
  int s = src[e], d = dst[e];
  wch[e] = -(dinvc[s] * dinvc[d]);   // scaled Laplacian edge weight
  wgc[e] = dinvg[s] * dinvg[d];      // GCN sym norm
}

// ---------------- sparse scatter: out[dst] += w[e] * h[src] ----------------

__global__ void scatter_kernel(const int* __restrict__ src, const int* __restrict__ dst,
                               const float* __restrict__ w, const float* __restrict__ Hin,
                               float* __restrict__ Hout, int E, int C) {
  int t = blockIdx.x * blockDim.x + threadIdx.x;
  int per = C >> 2;
  if (t >= E * per) return;
  int e  = t / per;
  int cb = (t - e * per) << 2;
  int s = src[e], d = dst[e];
  float we = w[e];
  const float4 h = *(const float4*)(Hin + (size_t)s * C + cb);
  float* o = Hout + (size_t)d * C + cb;
  atomicAdd(o + 0, we * h.x);
  atomicAdd(o + 1, we * h.y);
  atomicAdd(o + 2, we * h.z);
  atomicAdd(o + 3, we * h.w);
}

// T = 2*T - X0  (Chebyshev recurrence)
__global__ void combine_kernel(float* __restrict__ T, const float* __restrict__ X0, int total4) {
  int t = blockIdx.x * blockDim.x + threadIdx.x;
  if (t >= total4) return;
  float4 a = ((float4*)T)[t];
  float4 b = ((const float4*)X0)[t];
  a.x = 2.f * a.x - b.x;  a.y = 2.f * a.y - b.y;
  a.z = 2.f * a.z - b.z;  a.w = 2.f * a.w - b.w;
  ((float4*)T)[t] = a;
}

// ---------------- WMMA f32 GEMM: Out[M,COUT] += A[M,K] @ B[K,COUT] ----------------
// Compile-time K/COUT => all loads are base pointer + constant IOFFSET (no per-iter
// address math). Each wave owns MT stacked 16x16 tiles (MT*16 rows x 16 cols): the B
// fragment is loaded once per K-step and reused MT times, and the MT independent
// accumulator chains hide the WMMA->WMMA RAW hazard (no dependent back-to-back WMMA).
// EXEC stays all-1s through the WMMA region (wave-uniform early-exit; per-lane
// predication only on the final stores).

template <int K, int COUT, int MT>
__global__ void __launch_bounds__(TPB)
gemm_wmma(const float* __restrict__ A, const float* __restrict__ B,
          float* __restrict__ Out, int M) {
  int wid  = (blockIdx.x * blockDim.x + threadIdx.x) >> 5;
  int lane = threadIdx.x & 31;
  constexpr int tilesN = COUT >> 4;
  int tilesM = (M + 16 * MT - 1) / (16 * MT);
  if (wid >= tilesM * tilesN) return;
  int m0 = (wid / tilesN) * (16 * MT);
  int n0 = (wid % tilesN) << 4;
  int half = lane >> 4;     // 0: K pair {0,1}; 1: K pair {2,3}
  int l    = lane & 15;

  const float* Ap[MT];
#pragma unroll
  for (int t = 0; t < MT; ++t) {
    int ar = m0 + t * 16 + l; if (ar > M - 1) ar = M - 1;   // clamp tail (loads only)
    Ap[t] = A + (size_t)ar * K + 2 * half;
  }
  const float* Bp = B + (size_t)(2 * half) * COUT + n0 + l;

  v8f c[MT];
#pragma unroll
  for (int t = 0; t < MT; ++t) {
#pragma unroll
    for (int r = 0; r < 8; ++r) {
      int cr = m0 + t * 16 + r + 8 * half; if (cr > M - 1) cr = M - 1;
      c[t][r] = Out[(size_t)cr * COUT + n0 + l];
    }
  }

#pragma unroll
  for (int k0 = 0; k0 < K; k0 += 4) {
    v2f b; b[0] = Bp[k0 * COUT]; b[1] = Bp[(k0 + 1) * COUT];
#pragma unroll
    for (int t = 0; t < MT; ++t) {
      v2f a; a[0] = Ap[t][k0]; a[1] = Ap[t][k0 + 1];
      c[t] = __builtin_amdgcn_wmma_f32_16x16x4_f32(false, a, false, b, (short)0, c[t], false, false);
    }
  }

#pragma unroll
  for (int t = 0; t < MT; ++t) {
#pragma unroll
    for (int r = 0; r < 8; ++r) {
      int cr = m0 + t * 16 + r + 8 * half;
      if (cr < M) Out[(size_t)cr * COUT + n0 + l] = c[t][r];
    }
  }
}

// tiny scalar fallback for unexpected shapes (never used for this reference)
__global__ void gemm_simple(const float* __restrict__ A, const float* __restrict__ B,
                            float* __restrict__ Out, int M, int K, int Cout) {
  int t = blockIdx.x * blockDim.x + threadIdx.x;
  if (t >= M * Cout) return;
  int i = t / Cout, j = t - i * Cout;
  float acc = Out[t];
  for (int k = 0; k < K; ++k) acc += A[(size_t)i * K + k] * B[(size_t)k * Cout + j];
  Out[t] = acc;
}

// ---------------- elementwise ----------------

__global__ void bias_relu_kernel(float* __restrict__ H, const float* __restrict__ b, int N, int C) {
  int t = blockIdx.x * blockDim.x + threadIdx.x;
  int total4 = N * (C >> 2);
  if (t >= total4) return;
  int cb = (t % (C >> 2)) << 2;
  float4 v  = ((float4*)H)[t];
  float4 bb = *(const float4*)(b + cb);
  v.x = fmaxf(v.x + bb.x, 0.f);  v.y = fmaxf(v.y + bb.y, 0.f);
  v.z = fmaxf(v.z + bb.z, 0.f);  v.w = fmaxf(v.w + bb.w, 0.f);
  ((float4*)H)[t] = v;
}

// out[i,:] += dinvg[i]^2 * hlin[i,:]   (GCN self loop)
__global__ void selfloop_kernel(float* __restrict__ Outp, const float* __restrict__ Hlin,
                                const float* __restrict__ dinvg, int N, int C) {
  int t = blockIdx.x * blockDim.x + threadIdx.x;
  int per = C >> 2;
  if (t >= N * per) return;
  int i = t / per;
  float w = dinvg[i] * dinvg[i];
  float4 h = ((const float4*)Hlin)[t];
  float4 o = ((float4*)Outp)[t];
  o.x += w * h.x; o.y += w * h.y; o.z += w * h.z; o.w += w * h.w;
  ((float4*)Outp)[t] = o;
}

// ---------------- attention pooling ----------------

__global__ void gate_kernel(const float* __restrict__ H, const float* __restrict__ wg,
                            const float* __restrict__ bg, float* __restrict__ logits,
                            int N, int C) {
  int wid  = (blockIdx.x * blockDim.x + threadIdx.x) >> 5;
  int lane = threadIdx.x & 31;
  if (wid >= N) return;
  float acc = 0.f;
  for (int c = lane; c < C; c += 32) acc += H[(size_t)wid * C + c] * wg[c];
  for (int off = 16; off; off >>= 1) acc += __shfl_xor(acc, off, 32);
  if (lane == 0) logits[wid] = acc + bg[0];
}

__device__ __forceinline__ unsigned fenc(float f) {
  unsigned b = __float_as_uint(f);
  return (b & 0x80000000u) ? ~b : (b | 0x80000000u);
}
__device__ __forceinline__ float fdec(unsigned k) {
  unsigned b = (k & 0x80000000u) ? (k & 0x7FFFFFFFu) : ~k;
  return __uint_as_float(b);
}

__global__ void segmax_kernel(const float* __restrict__ logits, const int* __restrict__ batch,
                              unsigned* __restrict__ gmax, int N) {
  int i = blockIdx.x * blockDim.x + threadIdx.x;
  if (i >= N) return;
  atomicMax(&gmax[batch[i]], fenc(logits[i]));
}

__global__ void expsum_kernel(float* __restrict__ logits, const int* __restrict__ batch,
                              const unsigned* __restrict__ gmax, float* __restrict__ gsum, int N) {
  int i = blockIdx.x * blockDim.x + threadIdx.x;
  if (i >= N) return;
  int g = batch[i];
  float e = expf(logits[i] - fdec(gmax[g]));
  logits[i] = e;
  atomicAdd(&gsum[g], e);
}

__global__ void alpha_kernel(float* __restrict__ e, const int* __restrict__ batch,
                             const float* __restrict__ gsum, int N) {
  int i = blockIdx.x * blockDim.x + threadIdx.x;
  if (i >= N) return;
  e[i] = e[i] / gsum[batch[i]];
}

__global__ void pool_scatter_kernel(const float* __restrict__ alpha, const int* __restrict__ batch,
                                    const float* __restrict__ H, float* __restrict__ pooled,
                                    int N, int C) {
  int t = blockIdx.x * blockDim.x + threadIdx.x;
  int per = C >> 2;
  if (t >= N * per) return;
  int i  = t / per;
  int cb = (t - i * per) << 2;
  int g  = batch[i];
  float a = alpha[i];
  const float4 h = *(const float4*)(H + (size_t)i * C + cb);
  float* o = pooled + (size_t)g * C + cb;
  atomicAdd(o + 0, a * h.x);
  atomicAdd(o + 1, a * h.y);
  atomicAdd(o + 2, a * h.z);
  atomicAdd(o + 3, a * h.w);
}

// ---------------- final FC + log_softmax (tiny: 64x512 @ 512x10) ----------------

__global__ void fc_lsm_kernel(const float* __restrict__ pooled, const float* __restrict__ Wfc,
                              const float* __restrict__ bfc, float* __restrict__ out,
                              int G, int C, int NCn) {
  int g = blockIdx.x * blockDim.x + threadIdx.x;
  if (g >= G) return;
  float lg[16];
  for (int j = 0; j < NCn; ++j) lg[j] = bfc[j];
  for (int c = 0; c < C; ++c) {
    float pv = pooled[(size_t)g * C + c];
    for (int j = 0; j < NCn; ++j) lg[j] += pv * Wfc[(size_t)c * NCn + j];
  }
  float m = lg[0];
  for (int j = 1; j < NCn; ++j) m = fmaxf(m, lg[j]);
  float s = 0.f;
  for (int j = 0; j < NCn; ++j) s += expf(lg[j] - m);
  float lse = m + logf(s);
  for (int j = 0; j < NCn; ++j) out[(size_t)g * NCn + j] = lg[j] - lse;
}

// ---------------- host ----------------

#define GEMM_MT 8

static void launch_gemm(int K, int Cout, const float* A, const float* B, float* Out,
                        int M, hipStream_t s) {
  long long waves = (long long)cdiv(M, 16 * GEMM_MT) * (Cout / 16);
  int blocks = cdiv(waves * 32, TPB);
  if (K == 128 && Cout == 128) {
    gemm_wmma<128, 128, GEMM_MT><<<blocks, TPB, 0, s>>>(A, B, Out, M);
  } else if (K == 128 && Cout == 256) {
    gemm_wmma<128, 256, GEMM_MT><<<blocks, TPB, 0, s>>>(A, B, Out, M);
  } else if (K == 256 && Cout == 512) {
    gemm_wmma<256, 512, GEMM_MT><<<blocks, TPB, 0, s>>>(A, B, Out, M);
  } else {
    gemm_simple<<<cdiv((long long)M * Cout, TPB), TPB, 0, s>>>(A, B, Out, M, K, Cout);
  }
}

extern "C" void kernel_launch(void* const* d_in, const int* in_sizes, int n_in,
                              void* d_out, int out_size, void* d_ws, size_t ws_size,
                              hipStream_t stream) {
  (void)n_in; (void)ws_size;
  const float* x     = (const float*)d_in[0];
  const int*   ei    = (const int*)d_in[1];
  const int*   batch = (const int*)d_in[2];
  const float* Wch   = (const float*)d_in[3];
  const float* bch   = (const float*)d_in[4];
  const float* Wg1   = (const float*)d_in[5];
  const float* bg1   = (const float*)d_in[6];
  const float* Wg2   = (const float*)d_in[7];
  const float* bg2   = (const float*)d_in[8];
  const float* wgate = (const float*)d_in[9];
  const float* bgate = (const float*)d_in[10];
  const float* Wfc   = (const float*)d_in[11];
  const float* bfc   = (const float*)d_in[12];

  const int N   = in_sizes[2];
  const int E   = in_sizes[1] / 2;
  const int F   = in_sizes[0] / N;
  const int C1  = in_sizes[4];
  const int C2  = in_sizes[6];
  const int C3  = in_sizes[8];
  const int K   = in_sizes[3] / (F * C1);
  const int NCn = in_sizes[12];
  const int G   = out_size / NCn;

  const int* src = ei;
  const int* dst = ei + E;

  // workspace carve-up (~84 MB of f32)
  float* p = (float*)d_ws;
  float* degc  = p; p += N;
  float* degg  = p; p += N;
  float* dinvc = p; p += N;
  float* dinvg = p; p += N;
  float* wch   = p; p += E;
  float* wgc   = p; p += E;
  float* bufB  = p; p += (size_t)N * F;
  float* bufC  = p; p += (size_t)N * F;
  float* bufD  = p; p += (size_t)N * F;
  float* h1    = p; p += (size_t)N * C1;
  float* lin1  = p; p += (size_t)N * C2;
  float* h2    = p; p += (size_t)N * C2;
  float* lin2  = p; p += (size_t)N * C3;
  float* h3    = p; p += (size_t)N * C3;
  float* nbuf  = p; p += N;            // gate logits -> e -> alpha
  unsigned* gmax = (unsigned*)p;       // G uints
  float* gsum  = p + G;                // G floats
  float* pooled = p + 2 * G;           // G*C3 floats

  // degrees + edge norms
  hipMemsetAsync(degc, 0, sizeof(float) * 2 * (size_t)N, stream);
  deg_kernel<<<cdiv(E, TPB), TPB, 0, stream>>>(src, dst, degc, degg, E);
  dinv_kernel<<<cdiv(N, TPB), TPB, 0, stream>>>(degc, degg, dinvc, dinvg, N);
  edgew_kernel<<<cdiv(E, TPB), TPB, 0, stream>>>(src, dst, dinvc, dinvg, wch, wgc, E);

  // ---- ChebConv: h1 = relu(sum_k Tk(x) @ W_cheb[k] + b) ----
  hipMemsetAsync(h1, 0, sizeof(float) * (size_t)N * C1, stream);
  // T1 = prop(x)
  hipMemsetAsync(bufB, 0, sizeof(float) * (size_t)N * F, stream);
  scatter_kernel<<<cdiv((long long)E * (F / 4), TPB), TPB, 0, stream>>>(src, dst, wch, x, bufB, E, F);
  launch_gemm(F, C1, x,    Wch,                      h1, N, stream);   // T0 @ W0
  launch_gemm(F, C1, bufB, Wch + (size_t)1 * F * C1, h1, N, stream);   // T1 @ W1

  const float* t0 = x;
  float* t1 = bufB;
  float* freebufs[2] = { bufC, bufD };
  int nfree = 2;
  for (int k = 2; k < K; ++k) {
    float* tgt = freebufs[--nfree];
    hipMemsetAsync(tgt, 0, sizeof(float) * (size_t)N * F, stream);
    scatter_kernel<<<cdiv((long long)E * (F / 4), TPB), TPB, 0, stream>>>(src, dst, wch, t1, tgt, E, F);
    combine_kernel<<<cdiv((long long)N * F / 4, TPB), TPB, 0, stream>>>(tgt, t0, N * F / 4);
    launch_gemm(F, C1, tgt, Wch + (size_t)k * F * C1, h1, N, stream);
    if (t0 != x) freebufs[nfree++] = (float*)t0;
    t0 = t1; t1 = tgt;
  }
  bias_relu_kernel<<<cdiv((long long)N * C1 / 4, TPB), TPB, 0, stream>>>(h1, bch, N, C1);

  // ---- GCNConv 1: h2 = relu(agg(h1 @ W_g1) + b) ----
  hipMemsetAsync(lin1, 0, sizeof(float) * (size_t)N * C2, stream);
  launch_gemm(C1, C2, h1, Wg1, lin1, N, stream);
  hipMemsetAsync(h2, 0, sizeof(float) * (size_t)N * C2, stream);
  scatter_kernel<<<cdiv((long long)E * (C2 / 4), TPB), TPB, 0, stream>>>(src, dst, wgc, lin1, h2, E, C2);
  selfloop_kernel<<<cdiv((long long)N * C2 / 4, TPB), TPB, 0, stream>>>(h2, lin1, dinvg, N, C2);
  bias_relu_kernel<<<cdiv((long long)N * C2 / 4, TPB), TPB, 0, stream>>>(h2, bg1, N, C2);

  // ---- GCNConv 2: h3 = relu(agg(h2 @ W_g2) + b) ----
  hipMemsetAsync(lin2, 0, sizeof(float) * (size_t)N * C3, stream);
  launch_gemm(C2, C3, h2, Wg2, lin2, N, stream);
  hipMemsetAsync(h3, 0, sizeof(float) * (size_t)N * C3, stream);
  scatter_kernel<<<cdiv((long long)E * (C3 / 4), TPB), TPB, 0, stream>>>(src, dst, wgc, lin2, h3, E, C3);
  selfloop_kernel<<<cdiv((long long)N * C3 / 4, TPB), TPB, 0, stream>>>(h3, lin2, dinvg, N, C3);
  bias_relu_kernel<<<cdiv((long long)N * C3 / 4, TPB), TPB, 0, stream>>>(h3, bg2, N, C3);

  // ---- attention pooling ----
  hipMemsetAsync(gmax, 0, sizeof(float) * (2 * (size_t)G + (size_t)G * C3), stream);
  gate_kernel<<<cdiv((long long)N * 32, TPB), TPB, 0, stream>>>(h3, wgate, bgate, nbuf, N, C3);
  segmax_kernel<<<cdiv(N, TPB), TPB, 0, stream>>>(nbuf, batch, gmax, N);
  expsum_kernel<<<cdiv(N, TPB), TPB, 0, stream>>>(nbuf, batch, gmax, gsum, N);
  alpha_kernel<<<cdiv(N, TPB), TPB, 0, stream>>>(nbuf, batch, gsum, N);
  pool_scatter_kernel<<<cdiv((long long)N * C3 / 4, TPB), TPB, 0, stream>>>(nbuf, batch, h3, pooled, N, C3);

  // ---- FC + log_softmax ----
  fc_lsm_kernel<<<cdiv(G, 64), 64, 0, stream>>>(pooled, Wfc, bfc, (float*)d_out, G, C3, NCn);
}